// Encoding_8246337208719
// MI455X (gfx1250) — compile-verified
//
#include <hip/hip_runtime.h>
#include <math.h>

// Problem constants from the reference: N=16384, K=32, D=128.
#define KC 32
#define DC 128

typedef __attribute__((ext_vector_type(2))) float v2f;
typedef __attribute__((ext_vector_type(8))) float v8f;

// ---------------------------------------------------------------------------
// Kernel 0: zero the accumulators (S, E0) and compute cnorm[k] = ||c_k||^2.
// Must run every call: ws is poisoned once and graph replays don't re-zero.
// ---------------------------------------------------------------------------
__global__ __launch_bounds__(256) void init_kernel(const float* __restrict__ codes,
                                                   float* __restrict__ cnorm,
                                                   float* __restrict__ S,
                                                   float* __restrict__ E0) {
    const int t = threadIdx.x;
    for (int i = t; i < KC * DC; i += 256) E0[i] = 0.0f;
    if (t < KC) {
        S[t] = 0.0f;
        float acc = 0.0f;
        const float* row = codes + t * DC;
        for (int d = 0; d < DC; ++d) acc = fmaf(row[d], row[d], acc);
        cnorm[t] = acc;
    }
}

// ---------------------------------------------------------------------------
// Kernel 1 (WMMA GEMM #1, fused softmax):
//   Q = X * C^T via V_WMMA_F32_16X16X4_F32, one wave per 16 rows of x,
//   two accumulators for the two k-tiles (k 0..15, 16..31).
//   ||x_n||^2 is accumulated from the A fragment as a by-product.
//   Then l2 = sqrt(xn + cn - 2Q), softmax over K=32 done entirely in-register
//   with half-wave butterflies, W written to ws, S[k] accumulated atomically.
// ---------------------------------------------------------------------------
__global__ __launch_bounds__(256) void dist_softmax_kernel(
    const float* __restrict__ x, const float* __restrict__ codes,
    const float* __restrict__ scale, const float* __restrict__ cnorm,
    float* __restrict__ W, float* __restrict__ S) {

    const int lane = threadIdx.x & 31;
    const int wave = threadIdx.x >> 5;
    const int n0   = (blockIdx.x * 8 + wave) * 16;   // 16 rows of x per wave
    const int col  = lane & 15;                      // M (A) / N (B,C,D) index
    const int hi   = lane >> 4;                      // lane half
    const int r0   = hi * 2;                         // K-dim offsets {0,1}|{2,3}
    const int r1   = hi * 2 + 1;

    const float* xrow  = x     + (size_t)(n0 + col) * DC;  // A row m = col
    const float* c0row = codes + (size_t)col * DC;         // B col k = col
    const float* c1row = codes + (size_t)(col + 16) * DC;  // B col k = col+16

    v8f q0 = {};  // dot(x_n, c_k), k in [0,16)
    v8f q1 = {};  // dot(x_n, c_k), k in [16,32)
    float xnp = 0.0f;

    for (int db = 0; db < DC; db += 4) {
        v2f a, b0, b1;
        a.x  = xrow[db + r0];   a.y  = xrow[db + r1];
        b0.x = c0row[db + r0];  b0.y = c0row[db + r1];
        b1.x = c1row[db + r0];  b1.y = c1row[db + r1];
        xnp = fmaf(a.x, a.x, xnp);
        xnp = fmaf(a.y, a.y, xnp);
        q0 = __builtin_amdgcn_wmma_f32_16x16x4_f32(false, a, false, b0,
                                                   (short)0, q0, false, false);
        q1 = __builtin_amdgcn_wmma_f32_16x16x4_f32(false, a, false, b1,
                                                   (short)0, q1, false, false);
    }

    // Row norm: lanes l and l^16 cover complementary d-offsets of row (l%16).
    const float xn_all = xnp + __shfl_xor(xnp, 16, 32);  // lane l: ||x_{n0+l%16}||^2

    const float s0  = scale[col],       s1  = scale[col + 16];
    const float cn0 = cnorm[col],       cn1 = cnorm[col + 16];
    float sacc0 = 0.0f, sacc1 = 0.0f;

#pragma unroll
    for (int v = 0; v < 8; ++v) {
        const int m = v + hi * 8;                 // C/D row for this lane half
        const float xn = __shfl(xn_all, m, 32);   // broadcast ||x_row||^2
        const float l2a = sqrtf(fmaxf(xn + cn0 - 2.0f * q0[v], 0.0f));
        const float l2b = sqrtf(fmaxf(xn + cn1 - 2.0f * q1[v], 0.0f));
        float t0 = l2a * s0;
        float t1 = l2b * s1;
        // softmax over K=32: row lives across this 16-lane half (t0 | t1)
        float tm = fmaxf(t0, t1);
        for (int mask = 1; mask < 16; mask <<= 1)
            tm = fmaxf(tm, __shfl_xor(tm, mask, 32));
        const float e0 = __expf(t0 - tm);
        const float e1 = __expf(t1 - tm);
        float es = e0 + e1;
        for (int mask = 1; mask < 16; mask <<= 1)
            es += __shfl_xor(es, mask, 32);
        const float inv = 1.0f / es;
        const float w0 = e0 * inv;
        const float w1 = e1 * inv;
        const int n = n0 + m;
        W[(size_t)n * KC + col]      = w0;
        W[(size_t)n * KC + col + 16] = w1;
        sacc0 += w0;
        sacc1 += w1;
    }
    atomicAdd(&S[col],      sacc0);
    atomicAdd(&S[col + 16], sacc1);
}

// ---------------------------------------------------------------------------
// Kernel 2 (WMMA GEMM #2): E0 = W^T * X  (32x128, reduce over N).
// 16 output tiles x 32 N-slices; each wave does 128 f32 WMMA steps over its
// slice and atomically accumulates its 16x16 partial into E0.
// ---------------------------------------------------------------------------
__global__ __launch_bounds__(256) void wt_x_kernel(const float* __restrict__ x,
                                                   const float* __restrict__ W,
                                                   float* __restrict__ E0,
                                                   int nPerSlice) {
    const int lane  = threadIdx.x & 31;
    const int wave  = threadIdx.x >> 5;
    const int gw    = blockIdx.x * 8 + wave;  // 0..511
    const int tile  = gw & 15;
    const int slice = gw >> 4;                // 0..31
    const int kb    = (tile & 1) * 16;
    const int db    = (tile >> 1) * 16;
    const int col   = lane & 15;
    const int hi    = lane >> 4;
    const int nA    = hi * 2;                 // K-dim (n) offsets {0,1}|{2,3}

    v8f acc = {};
    const int nStart = slice * nPerSlice;
    for (int n = nStart; n < nStart + nPerSlice; n += 4) {
        v2f a, b;
        // A = W^T : A[M=k][Kd=n-off] = W[n][k]
        a.x = W[(size_t)(n + nA)     * KC + kb + col];
        a.y = W[(size_t)(n + nA + 1) * KC + kb + col];
        // B = X   : B[Kd=n-off][N=d]
        b.x = x[(size_t)(n + nA)     * DC + db + col];
        b.y = x[(size_t)(n + nA + 1) * DC + db + col];
        acc = __builtin_amdgcn_wmma_f32_16x16x4_f32(false, a, false, b,
                                                    (short)0, acc, false, false);
    }
#pragma unroll
    for (int v = 0; v < 8; ++v) {
        const int k = kb + v + hi * 8;
        atomicAdd(&E0[(size_t)k * DC + db + col], acc[v]);
    }
}

// ---------------------------------------------------------------------------
// Kernel 3: E[k][d] = E0[k][d] - codes[k][d] * S[k]
// ---------------------------------------------------------------------------
__global__ __launch_bounds__(256) void finalize_kernel(const float* __restrict__ E0,
                                                       const float* __restrict__ codes,
                                                       const float* __restrict__ S,
                                                       float* __restrict__ out) {
    const int i = blockIdx.x * 256 + threadIdx.x;  // 0..4095
    const int k = i >> 7;                          // / DC
    out[i] = fmaf(-codes[i], S[k], E0[i]);
}

extern "C" void kernel_launch(void* const* d_in, const int* in_sizes, int n_in,
                              void* d_out, int out_size, void* d_ws, size_t ws_size,
                              hipStream_t stream) {
    const float* x     = (const float*)d_in[0];   // (1, N, 128)
    const float* codes = (const float*)d_in[1];   // (32, 128)
    const float* scale = (const float*)d_in[2];   // (32,)
    float* out = (float*)d_out;                   // (32, 128)

    const int N = in_sizes[0] / DC;               // 16384

    // ws layout (floats): W[N*32] | cnorm[32] | S[32] | E0[4096]
    float* W     = (float*)d_ws;
    float* cnorm = W + (size_t)N * KC;
    float* S     = cnorm + KC;
    float* E0    = S + KC;

    init_kernel<<<1, 256, 0, stream>>>(codes, cnorm, S, E0);

    // one wave per 16 rows, 8 waves per block
    const int blocks1 = N / (16 * 8);
    dist_softmax_kernel<<<blocks1, 256, 0, stream>>>(x, codes, scale, cnorm, W, S);

    // 16 tiles * 32 slices = 512 waves = 64 blocks of 8 waves
    const int nPerSlice = N / 32;
    wt_x_kernel<<<64, 256, 0, stream>>>(x, W, E0, nPerSlice);

    finalize_kernel<<<(KC * DC) / 256, 256, 0, stream>>>(E0, codes, S, out);
}